// DarkChannel_8375186227709
// MI455X (gfx1250) — compile-verified
//
#include <hip/hip_runtime.h>

// Dark channel prior, fixed shapes from the reference:
//   x:   [16, 3, 512, 512] f32
//   out: [16, 1, 512, 512] f32
// Separable 15x15 erosion with reflect padding:
//   pass1: channel-min + horizontal 15-tap min  -> tmp (d_ws, 16 MiB)
//   pass2: vertical 15-tap min                  -> d_out
//
// Memory-bound problem (no matmul => WMMA inapplicable: (min) is not (+,*)).
// CDNA5-specific path: async global->LDS row staging (ASYNCcnt) via
// global_load_async_to_lds_b128; clean float4 fallback otherwise.

#define B_   16
#define C_   3
#define H_   512
#define W_   512
#define PAD_ 7

typedef float f4  __attribute__((ext_vector_type(4)));
typedef int   v4i __attribute__((ext_vector_type(4)));

#if defined(__gfx1250__) && __has_builtin(__builtin_amdgcn_global_load_async_to_lds_b128)
#define USE_ASYNC_LDS 1
typedef __attribute__((address_space(1))) v4i* as1_v4i_ptr;
typedef __attribute__((address_space(3))) v4i* as3_v4i_ptr;
#endif

__device__ __forceinline__ int refl(int i, int n) {
  if (i < 0) i = -i;                 // jnp 'reflect': no edge repeat
  if (i >= n) i = 2 * n - 2 - i;
  return i;
}

// ---------------------------------------------------------------------------
// Pass 1: one block per (b, h) row. 128 threads, 4 pixels/thread.
// raw[c][w] staged in LDS (async path), cm[w] = channel min, then
// horizontal 15-tap erosion via suffix/prefix mins.
// ---------------------------------------------------------------------------
__global__ __launch_bounds__(128) void pass1_rowmin(const float* __restrict__ x,
                                                    float* __restrict__ tmp) {
  __shared__ __align__(16) float raw[C_][W_];
  __shared__ __align__(16) float cm[W_];

  const int row = blockIdx.x;        // 0 .. B_*H_-1
  const int b   = row >> 9;
  const int h   = row & (H_ - 1);
  const int t   = threadIdx.x;       // 0..127
  const int w0  = t * 4;

  const float* rowbase = x + (size_t)((b * C_) * H_ + h) * W_;

#pragma unroll
  for (int c = 0; c < C_; ++c) {
    const float* gp = rowbase + c * (H_ * W_) + w0;
    float*       lp = &raw[c][w0];
#ifdef USE_ASYNC_LDS
    // global_load_async_to_lds_b128: per-lane 16B copy, tracked by ASYNCcnt.
    __builtin_amdgcn_global_load_async_to_lds_b128(
        (as1_v4i_ptr)(v4i*)gp,
        (as3_v4i_ptr)(v4i*)lp,
        /*imm offset*/ 0, /*cpol*/ 0);
#else
    *(f4*)lp = *(const f4*)gp;
#endif
  }

#ifdef USE_ASYNC_LDS
#if __has_builtin(__builtin_amdgcn_s_wait_asynccnt)
  __builtin_amdgcn_s_wait_asynccnt(0);
#else
  asm volatile("s_wait_asynccnt 0" ::: "memory");
#endif
#endif
  __syncthreads();

  // channel-wise min -> cm[]
  {
    f4 r0 = *(const f4*)&raw[0][w0];
    f4 r1 = *(const f4*)&raw[1][w0];
    f4 r2 = *(const f4*)&raw[2][w0];
    f4 m;
    m.x = fminf(fminf(r0.x, r1.x), r2.x);
    m.y = fminf(fminf(r0.y, r1.y), r2.y);
    m.z = fminf(fminf(r0.z, r1.z), r2.z);
    m.w = fminf(fminf(r0.w, r1.w), r2.w);
    *(f4*)&cm[w0] = m;
  }
  __syncthreads();

  // horizontal 15-tap min for 4 consecutive outputs: need cm[w0-7 .. w0+10]
  float vals[18];
#pragma unroll
  for (int k = 0; k < 18; ++k) vals[k] = cm[refl(w0 - PAD_ + k, W_)];

  float suf[15];
  suf[14] = vals[14];
#pragma unroll
  for (int i = 13; i >= 0; --i) suf[i] = fminf(vals[i], suf[i + 1]);

  f4 o;
  o.x = suf[0];                      // min(vals[0..14])
  float p = vals[15];
  o.y = fminf(suf[1], p);            // min(vals[1..15])
  p = fminf(p, vals[16]);
  o.z = fminf(suf[2], p);            // min(vals[2..16])
  p = fminf(p, vals[17]);
  o.w = fminf(suf[3], p);            // min(vals[3..17])

  *(f4*)(tmp + (size_t)(b * H_ + h) * W_ + w0) = o;
}

// ---------------------------------------------------------------------------
// Pass 2: vertical 15-tap min. Block = 32x8 threads, tile = 32 cols x 64 rows.
// Each thread: one column, 8 consecutive output rows from 22 loaded rows
// (2.75 loads/output; rows shared across neighboring blocks hit the 192MB L2).
// ---------------------------------------------------------------------------
__global__ __launch_bounds__(256) void pass2_colmin(const float* __restrict__ tmp,
                                                    float* __restrict__ out) {
  const int tx = threadIdx.x & 31;
  const int ty = threadIdx.x >> 5;
  const int w  = blockIdx.x * 32 + tx;
  const int h0 = (blockIdx.y * 8 + ty) * 8;   // 8 rows per thread
  const int b  = blockIdx.z;

  const float* base = tmp + (size_t)(b * H_) * W_ + w;

  float vals[22];                    // rows h0-7 .. h0+14 (reflected)
#pragma unroll
  for (int k = 0; k < 22; ++k) vals[k] = base[refl(h0 - PAD_ + k, H_) * W_];

  float suf[15];
  suf[14] = vals[14];
#pragma unroll
  for (int i = 13; i >= 0; --i) suf[i] = fminf(vals[i], suf[i + 1]);

  float* ob = out + (size_t)(b * H_ + h0) * W_ + w;
  ob[0] = suf[0];                    // min(vals[0..14])
  float p = __builtin_inff();
#pragma unroll
  for (int i = 1; i < 8; ++i) {
    p = fminf(p, vals[14 + i]);      // p = min(vals[15 .. 14+i])
    ob[i * W_] = fminf(suf[i], p);   // min(vals[i .. i+14])
  }
}

// ---------------------------------------------------------------------------
extern "C" void kernel_launch(void* const* d_in, const int* in_sizes, int n_in,
                              void* d_out, int out_size, void* d_ws, size_t ws_size,
                              hipStream_t stream) {
  (void)in_sizes; (void)n_in; (void)out_size;
  const float* x   = (const float*)d_in[0];
  float*       out = (float*)d_out;
  float*       tmp = (float*)d_ws;   // needs 16*512*512*4 = 16 MiB; fully
                                     // rewritten by pass1 before pass2 reads.
  (void)ws_size;

  pass1_rowmin<<<dim3(B_ * H_), dim3(128), 0, stream>>>(x, tmp);
  pass2_colmin<<<dim3(W_ / 32, H_ / 64, B_), dim3(256), 0, stream>>>(tmp, out);
}